// GcnNet_15358803050744
// MI455X (gfx1250) — compile-verified
//
#include <hip/hip_runtime.h>

typedef float v2f __attribute__((ext_vector_type(2)));
typedef float v4f __attribute__((ext_vector_type(4)));
typedef float v8f __attribute__((ext_vector_type(8)));

#define WG_GEMM 128
constexpr int KK = 320;   // K dimension for both WMMA GEMMs (C1 = 320)

// ---------------- degree / norm prep ----------------

__global__ void k_init_deg(float* deg, int N) {
    int i = blockIdx.x * blockDim.x + threadIdx.x;
    if (i < N) deg[i] = 1.0f;                    // self-loop contributes 1
}

__global__ void k_count_deg(const int* __restrict__ ei, float* deg, int E) {
    int e = blockIdx.x * blockDim.x + threadIdx.x;
    if (e < E) atomicAdd(&deg[ei[E + e]], 1.0f); // col = ei[E + e]
}

// deg -> dinv (in place), and s[i] = dinv^2 * x[i]  (self-loop term of layer 1)
__global__ void k_dinv_sinit(float* dinv, float* s, const float* __restrict__ x, int N) {
    int i = blockIdx.x * blockDim.x + threadIdx.x;
    if (i < N) {
        float d = dinv[i];
        float r = (d > 0.0f) ? rsqrtf(d) : 0.0f;
        dinv[i] = r;
        s[i] = r * r * x[i];
    }
}

// s[col] += dinv[row]*dinv[col]*x[row]   (layer-1 propagation, factored scalar)
__global__ void k_scatter_scalar(const int* __restrict__ ei, const float* __restrict__ dinv,
                                 const float* __restrict__ x, float* s, int E) {
    int e = blockIdx.x * blockDim.x + threadIdx.x;
    if (e < E) {
        int r = ei[e];
        int c = ei[E + e];
        atomicAdd(&s[c], dinv[r] * dinv[c] * x[r]);
    }
}

// h1[i,c] = s[i]*W1[c] + b1[c]   (pre-relu; relu fused into next GEMM's A-load)
__global__ void k_expand_h1(const float* __restrict__ s, const float* __restrict__ W1,
                            const float* __restrict__ b1, float* __restrict__ h1,
                            int N, int C) {
    int i = blockIdx.x * blockDim.x + threadIdx.x;
    if (i < N * C) {
        int node = i / C;
        int c    = i - node * C;
        h1[i] = s[node] * W1[c] + b1[c];
    }
}

// ---------------- WMMA f32 GEMM:  C = relu(A) @ B ----------------
// A: M x KK row-major (pre-relu), B: KK x Nout row-major, C: M x Nout.
// Block = 128 threads (4 waves). Block computes a 16-row x 64-col tile group;
// each wave owns one 16x16 tile, K-loop of 4 via V_WMMA_F32_16X16X4_F32.
__global__ void __launch_bounds__(WG_GEMM)
k_gemm_relu_wmma(const float* __restrict__ A, const float* __restrict__ B,
                 float* __restrict__ C, int Nout) {
    __shared__ float Alds[16 * KK];              // 20 KB fp32 A panel

    const int m0 = blockIdx.x * 16;
    const float* Ap = A + (size_t)m0 * KK;       // panel is contiguous 16*KK floats

    // cooperative relu-load of A panel into LDS (float4, coalesced)
    for (int i = threadIdx.x * 4; i < 16 * KK; i += WG_GEMM * 4) {
        v4f v = *(const v4f*)(Ap + i);
        v.x = fmaxf(v.x, 0.0f);
        v.y = fmaxf(v.y, 0.0f);
        v.z = fmaxf(v.z, 0.0f);
        v.w = fmaxf(v.w, 0.0f);
        *(v4f*)(Alds + i) = v;
    }
    __syncthreads();

    const int lane = threadIdx.x & 31;
    const int wave = threadIdx.x >> 5;
    const int half = lane >> 4;                  // 0: K-pair {0,1}, 1: K-pair {2,3}
    const int lr   = lane & 15;
    const int n    = blockIdx.y * 64 + wave * 16 + lr;

    const float* Acol = Alds + lr * KK + 2 * half;        // A[row=lr][k + 2*half + {0,1}]
    const float* Bp   = B + (size_t)(2 * half) * Nout + n; // B[k + 2*half + {0,1}][col=n]

    v8f acc = {0.f, 0.f, 0.f, 0.f, 0.f, 0.f, 0.f, 0.f};

#pragma unroll 4
    for (int k = 0; k < KK; k += 4) {
        v2f a = *(const v2f*)(Acol + k);          // 8B-aligned LDS read
        v2f b;
        b.x = Bp[(size_t)k * Nout];
        b.y = Bp[(size_t)k * Nout + Nout];
        // D = A(16x4) * B(4x16) + C, fp32 exact
        acc = __builtin_amdgcn_wmma_f32_16x16x4_f32(
            /*neg_a=*/false, a, /*neg_b=*/false, b,
            /*c_mod=*/(short)0, acc, /*reuse_a=*/false, /*reuse_b=*/false);
    }

    // D layout: vgpr v holds row (v + 8*half), col = lr
    float* Cp = C + (size_t)(m0 + 8 * half) * Nout + n;
#pragma unroll
    for (int v = 0; v < 8; ++v)
        Cp[(size_t)v * Nout] = acc[v];
}

// ---------------- propagation (layers 2,3) ----------------

// out[i,c] = b[c] + dinv[i]^2 * hW[i,c]   (bias + self-loop term)
__global__ void k_prop_init(const float* __restrict__ hW, const float* __restrict__ dinv,
                            const float* __restrict__ b, float* __restrict__ out,
                            int N, int C) {
    int i = blockIdx.x * blockDim.x + threadIdx.x;
    if (i < N * C) {
        int node = i / C;
        int c    = i - node * C;
        float r  = dinv[node];
        out[i] = b[c] + r * r * hW[i];
    }
}

// out[col,:] += dinv[row]*dinv[col] * hW[row,:]
// blockDim = (64, 4): 64 lanes sweep C features (coalesced), 4 edges per block.
__global__ void __launch_bounds__(256)
k_prop_edges(const int* __restrict__ ei, const float* __restrict__ dinv,
             const float* __restrict__ hW, float* __restrict__ out, int E, int C) {
    int e = blockIdx.x * blockDim.y + threadIdx.y;
    if (e >= E) return;
    int r = ei[e];
    int c = ei[E + e];
    float nrm = dinv[r] * dinv[c];
    const float* src = hW  + (size_t)r * C;
    float*       dst = out + (size_t)c * C;
    for (int f = threadIdx.x; f < C; f += 64)
        atomicAdd(&dst[f], nrm * src[f]);
}

// ---------------- final FC: out[i] = relu(h[i,:]) . Wfc + bfc ----------------
__global__ void k_fc(const float* __restrict__ h, const float* __restrict__ Wfc,
                     const float* __restrict__ bfc, float* __restrict__ out,
                     int N, int C) {
    int i = blockIdx.x * blockDim.x + threadIdx.x;
    if (i >= N) return;
    float acc = bfc[0];
    const float* hp = h + (size_t)i * C;
    for (int c = 0; c < C; ++c)
        acc += fmaxf(hp[c], 0.0f) * Wfc[c];
    out[i] = acc;
}

// ---------------- launch ----------------

extern "C" void kernel_launch(void* const* d_in, const int* in_sizes, int n_in,
                              void* d_out, int out_size, void* d_ws, size_t ws_size,
                              hipStream_t stream) {
    (void)n_in; (void)out_size; (void)ws_size;

    const float* x   = (const float*)d_in[0];
    const int*   ei  = (const int*)  d_in[1];
    const float* W1  = (const float*)d_in[2];
    const float* b1  = (const float*)d_in[3];
    const float* W2  = (const float*)d_in[4];
    const float* b2  = (const float*)d_in[5];
    const float* W3  = (const float*)d_in[6];
    const float* b3  = (const float*)d_in[7];
    const float* Wfc = (const float*)d_in[8];
    const float* bfc = (const float*)d_in[9];
    float* out = (float*)d_out;

    const int N  = in_sizes[0];        // 100000
    const int E  = in_sizes[1] / 2;    // 3200000
    const int C1 = in_sizes[2];        // 320
    const int C3 = in_sizes[6] / C1;   // 64

    float* ws   = (float*)d_ws;
    float* dinv = ws;                          // [N]
    float* sbuf = ws + N;                      // [N]
    float* bufA = ws + 2 * (size_t)N;          // [N*C1]
    float* bufB = bufA + (size_t)N * C1;       // [N*C1]

    // degrees -> dinv, plus layer-1 scalar self-loop term
    k_init_deg     <<<(N + 255) / 256, 256, 0, stream>>>(dinv, N);
    k_count_deg    <<<(E + 255) / 256, 256, 0, stream>>>(ei, dinv, E);
    k_dinv_sinit   <<<(N + 255) / 256, 256, 0, stream>>>(dinv, sbuf, x, N);
    k_scatter_scalar<<<(E + 255) / 256, 256, 0, stream>>>(ei, dinv, x, sbuf, E);

    // layer 1: h1 (pre-relu) = s * W1 + b1   -> bufA
    k_expand_h1<<<(N * C1 + 255) / 256, 256, 0, stream>>>(sbuf, W1, b1, bufA, N, C1);

    // layer 2: hW2 = relu(h1) @ W2 -> bufB   (WMMA fp32)
    {
        dim3 g(N / 16, C1 / 64), b(WG_GEMM);
        k_gemm_relu_wmma<<<g, b, 0, stream>>>(bufA, W2, bufB, C1);
    }
    // layer-2 propagation: bufA = b2 + dinv^2*hW2, then edge atomics
    k_prop_init<<<(N * C1 + 255) / 256, 256, 0, stream>>>(bufB, dinv, b2, bufA, N, C1);
    {
        dim3 g((E + 3) / 4), b(64, 4);
        k_prop_edges<<<g, b, 0, stream>>>(ei, dinv, bufB, bufA, E, C1);
    }

    // layer 3: hW3 = relu(h2) @ W3 -> bufB   (WMMA fp32, Nout = 64)
    {
        dim3 g(N / 16, C3 / 64), b(WG_GEMM);
        k_gemm_relu_wmma<<<g, b, 0, stream>>>(bufA, W3, bufB, C3);
    }
    // layer-3 propagation -> bufA (h2 dead)
    k_prop_init<<<(N * C3 + 255) / 256, 256, 0, stream>>>(bufB, dinv, b3, bufA, N, C3);
    {
        dim3 g((E + 3) / 4), b(64, 4);
        k_prop_edges<<<g, b, 0, stream>>>(ei, dinv, bufB, bufA, E, C3);
    }

    // final FC
    k_fc<<<(N + 255) / 256, 256, 0, stream>>>(bufA, Wfc, bfc, out, N, C3);
}